// SelfConditioningResidualLayer_72533407694861
// MI455X (gfx1250) — compile-verified
//
#include <hip/hip_runtime.h>
#include <hip/hip_bf16.h>
#include <math.h>

typedef __attribute__((ext_vector_type(2))) float v2f;
typedef __attribute__((ext_vector_type(8))) float v8f;

#define NN        50000
#define EE        1000000
#define E_HALF    500000
#define NODE_DIM  256
#define EDGE_DIM  64
#define RBF_DIM   32
#define N_CAT_A   16
#define N_CAT_E   5
#define KPAD      104          // 101 features padded to multiple of 4
#define INV_SIGMA 3.2f         // 1/(10/32)

#define NODE_ROWS 48           // nodes per block (3 M-tiles of 16)
#define EDGE_ROWS 128          // edges per block (4 M-tiles of 16 per wave-half)

__device__ __forceinline__ float rbf_mu(int k) {
    return 10.0f * (float)k * (1.0f / 31.0f);   // linspace(0,10,32)
}
__device__ __forceinline__ float silu(float x) {
    return x / (1.0f + __expf(-x));
}
__device__ __forceinline__ v8f splat8(float v) {
    v8f r = {v, v, v, v, v, v, v, v};
    return r;
}
__device__ __forceinline__ v8f wmma_f32(v2f a, v2f b, v8f c) {
    return __builtin_amdgcn_wmma_f32_16x16x4_f32(false, a, false, b, (short)0, c, false, false);
}

// ---------------- Node kernel: 48 nodes per 256-thread block (8 waves) --------
// Three 16-row M-tiles per wave share every B fragment (W_n L2 traffic /3).
__global__ __launch_bounds__(256) void node_kernel(
    const float* __restrict__ s_t,  const float* __restrict__ x_t,
    const float* __restrict__ v_t,  const float* __restrict__ dst_x,
    const float* __restrict__ dst_v,const float* __restrict__ dst_a,
    const float* __restrict__ W_x,  const float* __restrict__ b_x,
    const float* __restrict__ W_v,  const float* __restrict__ b_v,
    const float* __restrict__ W_a,  const float* __restrict__ b_a,
    const float* __restrict__ W_n,  const float* __restrict__ b_n,
    float* __restrict__ node_out)
{
    __shared__ float feat[NODE_ROWS][68];    // 48x64 features, padded stride
    __shared__ float rbuf[NODE_ROWS][260];   // 48x256 intermediate, padded stride

    const int node0 = blockIdx.x * NODE_ROWS;
    const int tid   = threadIdx.x;
    const int lane  = tid & 31;
    const int wave  = tid >> 5;

    // ---- build feature tile: [rbf(32) | d_v(16) | dst_a(16)] ----
    for (int idx = tid; idx < NODE_ROWS * 64; idx += 256) {
        const int m = idx >> 6, k = idx & 63;
        int g = node0 + m; if (g >= NN) g = NN - 1;   // clamp: garbage rows never stored
        float val;
        if (k < 32) {
            float dx = x_t[g*3+0] - dst_x[g*3+0];
            float dy = x_t[g*3+1] - dst_x[g*3+1];
            float dz = x_t[g*3+2] - dst_x[g*3+2];
            float d  = sqrtf(fmaxf(dx*dx + dy*dy + dz*dz, 1e-8f));
            float t  = (d - rbf_mu(k)) * INV_SIGMA;
            val = __expf(-t * t);
        } else if (k < 48) {
            const int c1 = (k - 32) >> 2, c2 = (k - 32) & 3;
            float dx = v_t[(g*4+c1)*3+0] - dst_v[(g*4+c2)*3+0];
            float dy = v_t[(g*4+c1)*3+1] - dst_v[(g*4+c2)*3+1];
            float dz = v_t[(g*4+c1)*3+2] - dst_v[(g*4+c2)*3+2];
            val = sqrtf(fmaxf(dx*dx + dy*dy + dz*dz, 1e-8f));
        } else {
            val = dst_a[g*16 + (k - 48)];
        }
        feat[m][k] = val;
    }
    __syncthreads();

    const int mlo   = lane & 15;            // row within M-tile
    const int khalf = (lane < 16) ? 0 : 2;
    const int hi8   = (lane < 16) ? 0 : 8;  // C/D row offset for high half-wave

    // ---- GEMM1: r = feat(48x64) @ [W_x;W_v;W_a](64x256) + bias ----
    for (int t = 0; t < 2; ++t) {
        const int colbase = (wave + t * 8) * 16;
        const int n = colbase + mlo;
        const float bias = b_x[n] + b_v[n] + b_a[n];
        v8f acc[3];
        #pragma unroll
        for (int j = 0; j < 3; ++j) acc[j] = splat8(bias);
        #pragma unroll
        for (int i = 0; i < 8; ++i) {            // k 0..31 -> W_x
            const int k = 4*i + khalf;
            v2f b = { W_x[(size_t)k*NODE_DIM + n], W_x[(size_t)(k+1)*NODE_DIM + n] };
            #pragma unroll
            for (int j = 0; j < 3; ++j) {
                v2f a = { feat[16*j + mlo][k], feat[16*j + mlo][k+1] };
                acc[j] = wmma_f32(a, b, acc[j]);
            }
        }
        #pragma unroll
        for (int i = 0; i < 4; ++i) {            // k 32..47 -> W_v
            const int k = 4*i + khalf;
            v2f b = { W_v[(size_t)k*NODE_DIM + n], W_v[(size_t)(k+1)*NODE_DIM + n] };
            #pragma unroll
            for (int j = 0; j < 3; ++j) {
                v2f a = { feat[16*j + mlo][32+k], feat[16*j + mlo][32+k+1] };
                acc[j] = wmma_f32(a, b, acc[j]);
            }
        }
        #pragma unroll
        for (int i = 0; i < 4; ++i) {            // k 48..63 -> W_a
            const int k = 4*i + khalf;
            v2f b = { W_a[(size_t)k*NODE_DIM + n], W_a[(size_t)(k+1)*NODE_DIM + n] };
            #pragma unroll
            for (int j = 0; j < 3; ++j) {
                v2f a = { feat[16*j + mlo][48+k], feat[16*j + mlo][48+k+1] };
                acc[j] = wmma_f32(a, b, acc[j]);
            }
        }
        #pragma unroll
        for (int j = 0; j < 3; ++j)
            #pragma unroll
            for (int i = 0; i < 8; ++i)
                rbuf[16*j + i + hi8][colbase + mlo] = acc[j][i];
    }
    __syncthreads();

    // ---- GEMM2: out = silu(r(48x256) @ W_n(256x256) + b_n) + s_t ----
    for (int t = 0; t < 2; ++t) {
        const int colbase = (wave + t * 8) * 16;
        const int n = colbase + mlo;
        const float bias = b_n[n];
        v8f acc[3];
        #pragma unroll
        for (int j = 0; j < 3; ++j) acc[j] = splat8(bias);
        #pragma unroll 4
        for (int i = 0; i < 64; ++i) {
            const int k = 4*i + khalf;
            v2f b = { W_n[(size_t)k*NODE_DIM + n], W_n[(size_t)(k+1)*NODE_DIM + n] };
            #pragma unroll
            for (int j = 0; j < 3; ++j) {
                v2f a = { rbuf[16*j + mlo][k], rbuf[16*j + mlo][k+1] };
                acc[j] = wmma_f32(a, b, acc[j]);
            }
        }
        #pragma unroll
        for (int j = 0; j < 3; ++j) {
            #pragma unroll
            for (int i = 0; i < 8; ++i) {
                const int r = 16*j + i + hi8;
                if (node0 + r < NN) {
                    const size_t off = (size_t)(node0 + r) * NODE_DIM + n;
                    node_out[off] = s_t[off] + silu(acc[j][i]);
                }
            }
        }
    }
}

// ---------------- Edge kernel: 128 edges per 256-thread block (8 waves) ------
// Wave w: column tile (w&3), four M-tiles at base (w>=4)?64:0.
__global__ __launch_bounds__(256) void edge_kernel(
    const float* __restrict__ x_t,   const float* __restrict__ e_t,
    const float* __restrict__ dst_x, const float* __restrict__ dst_e,
    const int*   __restrict__ edge_src, const int* __restrict__ edge_dst,
    const float* __restrict__ W_e,   const float* __restrict__ b_e,
    float* __restrict__ edge_out)
{
    __shared__ float feat[EDGE_ROWS][105];   // 128 x 104 (101 + pad), padded stride
    __shared__ float dist_t[EDGE_ROWS];
    __shared__ float dist_1[EDGE_ROWS];

    const int e0   = blockIdx.x * EDGE_ROWS;
    const int tid  = threadIdx.x;
    const int lane = tid & 31;
    const int wave = tid >> 5;

    if (tid < EDGE_ROWS) {
        int e = e0 + tid; if (e >= E_HALF) e = E_HALF - 1;
        const int s = edge_src[e], d = edge_dst[e];
        float dx = x_t[s*3+0] - x_t[d*3+0];
        float dy = x_t[s*3+1] - x_t[d*3+1];
        float dz = x_t[s*3+2] - x_t[d*3+2];
        dist_t[tid] = sqrtf(fmaxf(dx*dx + dy*dy + dz*dz, 1e-8f));
        dx = dst_x[s*3+0] - dst_x[d*3+0];
        dy = dst_x[s*3+1] - dst_x[d*3+1];
        dz = dst_x[s*3+2] - dst_x[d*3+2];
        dist_1[tid] = sqrtf(fmaxf(dx*dx + dy*dy + dz*dz, 1e-8f));
    }
    __syncthreads();

    // features: [e_t(64) | dst_e(5) | rbf(d1)-rbf(dt)(32) | pad(3)]
    for (int idx = tid; idx < EDGE_ROWS * KPAD; idx += 256) {
        const int r = idx / KPAD;
        const int k = idx - r * KPAD;
        int e = e0 + r; if (e >= E_HALF) e = E_HALF - 1;
        float val;
        if (k < 64) {
            val = e_t[(size_t)e * EDGE_DIM + k];
        } else if (k < 69) {
            val = dst_e[(size_t)e * N_CAT_E + (k - 64)];
        } else if (k < 101) {
            const int j = k - 69;
            const float mu = rbf_mu(j);
            float t1 = (dist_1[r] - mu) * INV_SIGMA;
            float t0 = (dist_t[r] - mu) * INV_SIGMA;
            val = __expf(-t1 * t1) - __expf(-t0 * t0);
        } else {
            val = 0.0f;
        }
        feat[r][k] = val;
    }
    __syncthreads();

    // GEMM: (128x104) @ W_e(101x64); B fragments shared across four M-tiles
    const int mbase   = (wave >= 4) ? 64 : 0;
    const int colbase = (wave & 3) * 16;
    const int mlo   = lane & 15;
    const int n     = colbase + mlo;
    const int khalf = (lane < 16) ? 0 : 2;
    const int hi8   = (lane < 16) ? 0 : 8;

    const float bias = b_e[n];
    v8f acc[4];
    #pragma unroll
    for (int j = 0; j < 4; ++j) acc[j] = splat8(bias);
    #pragma unroll 2
    for (int i = 0; i < 26; ++i) {
        const int k  = 4*i + khalf;
        const int kb0 = (k     > 100) ? 100 : k;       // clamp: A is zero there
        const int kb1 = (k + 1 > 100) ? 100 : k + 1;
        v2f b = { W_e[(size_t)kb0 * EDGE_DIM + n], W_e[(size_t)kb1 * EDGE_DIM + n] };
        #pragma unroll
        for (int j = 0; j < 4; ++j) {
            v2f a = { feat[mbase + 16*j + mlo][k], feat[mbase + 16*j + mlo][k+1] };
            acc[j] = wmma_f32(a, b, acc[j]);
        }
    }
    #pragma unroll
    for (int j = 0; j < 4; ++j) {
        #pragma unroll
        for (int i = 0; i < 8; ++i) {
            const int row = mbase + 16*j + i + hi8;
            const int e = e0 + row;
            if (e < E_HALF) {
                const float y = silu(acc[j][i]);
                edge_out[(size_t)e * EDGE_DIM + n] = y;                 // upper half
                edge_out[(size_t)(e + E_HALF) * EDGE_DIM + n] = y;      // mirrored
            }
        }
    }
}

extern "C" void kernel_launch(void* const* d_in, const int* in_sizes, int n_in,
                              void* d_out, int out_size, void* d_ws, size_t ws_size,
                              hipStream_t stream) {
    const float* s_t   = (const float*)d_in[0];
    const float* x_t   = (const float*)d_in[1];
    const float* v_t   = (const float*)d_in[2];
    const float* e_t   = (const float*)d_in[3];
    const float* dst_x = (const float*)d_in[4];
    const float* dst_v = (const float*)d_in[5];
    const float* dst_a = (const float*)d_in[6];
    const float* dst_e = (const float*)d_in[7];
    const int*   esrc  = (const int*)d_in[8];
    const int*   edst  = (const int*)d_in[9];
    const float* W_x = (const float*)d_in[10]; const float* b_x = (const float*)d_in[11];
    const float* W_v = (const float*)d_in[12]; const float* b_v = (const float*)d_in[13];
    const float* W_a = (const float*)d_in[14]; const float* b_a = (const float*)d_in[15];
    const float* W_n = (const float*)d_in[16]; const float* b_n = (const float*)d_in[17];
    const float* W_e = (const float*)d_in[18]; const float* b_e = (const float*)d_in[19];

    float* node_out = (float*)d_out;                         // [N, 256]
    float* edge_out = (float*)d_out + (size_t)NN * NODE_DIM; // [E, 64]

    node_kernel<<<(NN + NODE_ROWS - 1) / NODE_ROWS, 256, 0, stream>>>(
        s_t, x_t, v_t, dst_x, dst_v, dst_a,
        W_x, b_x, W_v, b_v, W_a, b_a, W_n, b_n, node_out);

    edge_kernel<<<(E_HALF + EDGE_ROWS - 1) / EDGE_ROWS, 256, 0, stream>>>(
        x_t, e_t, dst_x, dst_e, esrc, edst, W_e, b_e, edge_out);
}